// LDM_79774722555962
// MI455X (gfx1250) — compile-verified
//
#include <hip/hip_runtime.h>
#include <hip/hip_bf16.h>

// Problem constants (from reference setup_inputs)
#define BB 128   // batch (nseq)
#define TT 128   // time steps
#define DX 64    // state dim
#define DA 96    // observation dim
#define DU 16    // input dim
#define NTHR 256 // 8 waves of 32

typedef __attribute__((ext_vector_type(2))) float v2f;
typedef __attribute__((ext_vector_type(8))) float v8f;

// ---------------- LDS layout (floats) ----------------
constexpr int OFF_C    = 0;                    // C   [DA][DX]
constexpr int OFF_A    = OFF_C    + DA * DX;   // A   [DX][DX]
constexpr int OFF_B    = OFF_A    + DX * DX;   // Bm  [DX][DU]
constexpr int OFF_LAM  = OFF_B    + DX * DU;   // Lam_pred [DX][DX]
constexpr int OFF_LAMT = OFF_LAM  + DX * DX;   // Lam_t    [DX][DX]
constexpr int OFF_M1   = OFF_LAMT + DX * DX;   // M1 = Lam@C^T [DX][DA] (reused: -KC [DX][DX])
constexpr int OFF_S    = OFF_M1   + DX * DA;   // S   [DA][DA]
constexpr int OFF_INV  = OFF_S    + DA * DA;   // S^-1 [DA][DA]
constexpr int OFF_K    = OFF_INV  + DA * DA;   // K   [DX][DA] (reused: A@Lam_t [DX][DX])
constexpr int OFF_W    = OFF_K    + DX * DA;   // W diag [DX]
constexpr int OFF_R    = OFF_W    + DX;        // R diag [DA]
constexpr int OFF_MU   = OFF_R    + DA;        // mu_pred [DX]
constexpr int OFF_MUT  = OFF_MU   + DX;        // mu_t    [DX]
constexpr int OFF_RV   = OFF_MUT  + DX;        // innovation r [DA]
constexpr int OFF_F    = OFF_RV   + DA;        // GJ elimination factors [DA]
constexpr int OFF_PIV  = OFF_F    + DA;        // GJ pivot reciprocal [1]
constexpr int SMEM_FLOATS = OFF_PIV + 1;       // ~50657 floats ~= 203 KB (fits 320 KB/WGP)

// ---------------- WMMA helpers (fp32, 16x16x4) ----------------
__device__ __forceinline__ v8f wmma_f32(v2f a, v2f b, v8f c) {
    // (neg_a, A, neg_b, B, c_mod, C, reuse_a, reuse_b)
    return __builtin_amdgcn_wmma_f32_16x16x4_f32(false, a, false, b, (short)0, c,
                                                 false, false);
}

// Accumulate one 16x16 tile of D = opA(Am) * opB(Bm) into acc.
// TA: A stored transposed (A[m][k] = Am[k*lda + m]); TB: B stored transposed
// (B[k][n] = Bm[n*ldb + k], i.e. Bm is row-major N x K).
template <bool TA, bool TB>
__device__ __forceinline__ v8f tile_mm(const float* __restrict__ Am, int lda,
                                       const float* __restrict__ Bm, int ldb,
                                       int tm, int tn, int K, v8f acc, int lane) {
    const int r  = lane & 15;          // row-in-tile for A frag, col-in-tile for B frag
    const int kh = (lane >> 4) << 1;   // K sub-offset: lanes 0-15 -> K 0..1, 16-31 -> K 2..3
    const int am = tm * 16 + r;
    const int bn = tn * 16 + r;
    for (int k0 = 0; k0 < K; k0 += 4) {
        const int ka = k0 + kh;
        v2f af, bf;
        af.x = TA ? Am[(ka + 0) * lda + am] : Am[am * lda + (ka + 0)];
        af.y = TA ? Am[(ka + 1) * lda + am] : Am[am * lda + (ka + 1)];
        bf.x = TB ? Bm[bn * ldb + (ka + 0)] : Bm[(ka + 0) * ldb + bn];
        bf.y = TB ? Bm[bn * ldb + (ka + 1)] : Bm[(ka + 1) * ldb + bn];
        acc = wmma_f32(af, bf, acc);
    }
    return acc;
}

__device__ __forceinline__ void tile_store(float* __restrict__ D, int ldd,
                                           int tm, int tn, v8f acc, int lane, float sgn) {
    const int n  = tn * 16 + (lane & 15);
    const int m0 = tm * 16 + ((lane >> 4) << 3);  // lanes 16-31 hold rows M+8
#pragma unroll
    for (int i = 0; i < 8; ++i) D[(m0 + i) * ldd + n] = sgn * acc[i];
}

__device__ __forceinline__ v8f tile_load(const float* __restrict__ S, int lds_,
                                         int tm, int tn, int lane) {
    const int n  = tn * 16 + (lane & 15);
    const int m0 = tm * 16 + ((lane >> 4) << 3);
    v8f acc;
#pragma unroll
    for (int i = 0; i < 8; ++i) acc[i] = S[(m0 + i) * lds_ + n];
    return acc;
}

// D(MxN) = sgn * opA(Am) * opB(Bm) [+ Cinit]; tiles distributed over 8 waves.
template <bool TA, bool TB>
__device__ __forceinline__ void mm(float* __restrict__ D, int ldd,
                                   const float* __restrict__ Am, int lda,
                                   const float* __restrict__ Bm, int ldb,
                                   int M, int N, int K,
                                   const float* __restrict__ Cinit, int ldc,
                                   float sgn, int wave, int lane) {
    const int tN = N >> 4;
    const int nt = (M >> 4) * tN;
    for (int t = wave; t < nt; t += 8) {
        const int tm = t / tN, tn = t - tm * tN;
        v8f acc;
        if (Cinit) acc = tile_load(Cinit, ldc, tm, tn, lane);
        else       acc = (v8f){0.f, 0.f, 0.f, 0.f, 0.f, 0.f, 0.f, 0.f};
        acc = tile_mm<TA, TB>(Am, lda, Bm, ldb, tm, tn, K, acc, lane);
        tile_store(D, ldd, tm, tn, acc, lane, sgn);
    }
}

// ---------------- Kernel: one block per sequence b ----------------
__global__ void __launch_bounds__(NTHR, 1)
ldm_kalman_fwd(const float* __restrict__ g_a,    // (B,T,DA)
               const float* __restrict__ g_u,    // (B,T,DU)
               const float* __restrict__ g_mask, // (B,T)
               const float* __restrict__ g_A,    // (DX,DX)
               const float* __restrict__ g_Bm,   // (DX,DU)
               const float* __restrict__ g_C,    // (DA,DX)
               const float* __restrict__ g_mu0,  // (DX)
               const float* __restrict__ g_Lam0, // (DX,DX)
               const float* __restrict__ g_Wlog, // (DX)
               const float* __restrict__ g_Rlog, // (DA)
               float* __restrict__ out) {
    extern __shared__ float smem[];
    float* sC    = smem + OFF_C;
    float* sA    = smem + OFF_A;
    float* sBm   = smem + OFF_B;
    float* sLam  = smem + OFF_LAM;
    float* sLamT = smem + OFF_LAMT;
    float* sM1   = smem + OFF_M1;
    float* sS    = smem + OFF_S;
    float* sInv  = smem + OFF_INV;
    float* sK    = smem + OFF_K;
    float* sW    = smem + OFF_W;
    float* sR    = smem + OFF_R;
    float* sMu   = smem + OFF_MU;
    float* sMuT  = smem + OFF_MUT;
    float* sRv   = smem + OFF_RV;
    float* sF    = smem + OFF_F;
    float* sPiv  = smem + OFF_PIV;

    const int b    = blockIdx.x;
    const int tid  = threadIdx.x;
    const int wave = tid >> 5;
    const int lane = tid & 31;

    // ---- one-time setup: stage all model parameters + initial state in LDS ----
    for (int i = tid; i < DA * DX; i += NTHR) sC[i] = g_C[i];
    for (int i = tid; i < DX * DX; i += NTHR) { sA[i] = g_A[i]; sLam[i] = g_Lam0[i]; }
    for (int i = tid; i < DX * DU; i += NTHR) sBm[i] = g_Bm[i];
    if (tid < DX) { sW[tid] = __expf(g_Wlog[tid]); sMu[tid] = g_mu0[tid]; }
    if (tid < DA) sR[tid] = __expf(g_Rlog[tid]);
    __syncthreads();

    float* out_mu_next  = out;                                    // (T,B,DX)
    float* out_mu_t     = out + (size_t)TT * BB * DX;             // (T,B,DX)
    float* out_Lam_next = out + (size_t)2 * TT * BB * DX;         // (T,B,DX,DX)
    float* out_Lam_t    = out_Lam_next + (size_t)TT * BB * DX * DX;

#pragma unroll 1
    for (int t = 0; t < TT; ++t) {
        const float m_t = g_mask[(size_t)b * TT + t];

        // innovation r = m*a_t - C @ mu_pred  (runs alongside WMMA for M1)
        if (tid < DA) {
            float acc = 0.f;
            for (int x = 0; x < DX; ++x) acc += sC[tid * DX + x] * sMu[x];
            sRv[tid] = g_a[((size_t)b * TT + t) * DA + tid] * m_t - acc;
        }
        // M1 = Lam_pred @ C^T   (64x96)
        mm<false, true>(sM1, DA, sLam, DX, sC, DX, DX, DA, DX,
                        (const float*)nullptr, 0, 1.f, wave, lane);
        __syncthreads();

        // S = C @ M1            (96x96); zero S^-1 concurrently
        mm<false, false>(sS, DA, sC, DX, sM1, DA, DA, DA, DX,
                         (const float*)nullptr, 0, 1.f, wave, lane);
        for (int i = tid; i < DA * DA; i += NTHR) sInv[i] = 0.f;
        __syncthreads();
        if (tid < DA) { sS[tid * DA + tid] += sR[tid]; sInv[tid * DA + tid] = 1.f; }
        __syncthreads();

        // Gauss-Jordan inversion of S (SPD, no pivoting) into sInv
#pragma unroll 1
        for (int k = 0; k < DA; ++k) {
            if (tid == 0) sPiv[0] = 1.f / sS[k * DA + k];
            __syncthreads();
            const float ip = sPiv[0];
            for (int j = tid; j < 2 * DA; j += NTHR) {
                if (j < DA) sS[k * DA + j] *= ip;
                else        sInv[k * DA + (j - DA)] *= ip;
            }
            if (tid < DA) sF[tid] = (tid == k) ? 0.f : sS[tid * DA + k];
            __syncthreads();
            for (int idx = tid; idx < DA * 2 * DA; idx += NTHR) {
                const int i = idx / (2 * DA);
                const int j = idx - i * (2 * DA);
                if (i == k) continue;
                const float f = sF[i];
                if (j < DA) sS[i * DA + j]        -= f * sS[k * DA + j];
                else        sInv[i * DA + (j - DA)] -= f * sInv[k * DA + (j - DA)];
            }
            __syncthreads();
        }

        // K = M1 @ S^-1   (64x96), then mask
        mm<false, false>(sK, DA, sM1, DA, sInv, DA, DX, DA, DA,
                         (const float*)nullptr, 0, 1.f, wave, lane);
        __syncthreads();
        for (int i = tid; i < DX * DA; i += NTHR) sK[i] *= m_t;
        __syncthreads();

        // mu_t = mu_pred + K @ r  (runs alongside WMMA for -KC)
        if (tid < DX) {
            float acc = sMu[tid];
            for (int j = 0; j < DA; ++j) acc += sK[tid * DA + j] * sRv[j];
            sMuT[tid] = acc;
        }
        // -KC = -(K @ C)  (64x64) into sM1 (M1 dead now)
        mm<false, false>(sM1, DX, sK, DA, sC, DX, DX, DX, DA,
                         (const float*)nullptr, 0, -1.f, wave, lane);
        __syncthreads();

        // Lam_t = Lam_pred + (-KC) @ Lam_pred  (acc seeded with Lam_pred tile)
        mm<false, false>(sLamT, DX, sM1, DX, sLam, DX, DX, DX, DX,
                         sLam, DX, 1.f, wave, lane);
        __syncthreads();

        // Tmp = A @ Lam_t  into sK (K dead now)
        mm<false, false>(sK, DX, sA, DX, sLamT, DX, DX, DX, DX,
                         (const float*)nullptr, 0, 1.f, wave, lane);
        __syncthreads();

        // Lam_next = Tmp @ A^T  into sLam (Lam_pred dead now)
        mm<false, true>(sLam, DX, sK, DX, sA, DX, DX, DX, DX,
                        (const float*)nullptr, 0, 1.f, wave, lane);
        __syncthreads();

        // diag(W) add; mu_next = A@mu_t + Bm@u_t; write mu outputs
        if (tid < DX) {
            sLam[tid * DX + tid] += sW[tid];
            float acc = 0.f;
            for (int x = 0; x < DX; ++x) acc += sA[tid * DX + x] * sMuT[x];
            const float* ut = g_u + ((size_t)b * TT + t) * DU;
            for (int j = 0; j < DU; ++j) acc += sBm[tid * DU + j] * ut[j];
            sMu[tid] = acc;  // becomes next mu_pred
            const size_t o = ((size_t)t * BB + b) * DX + tid;
            out_mu_next[o] = acc;
            out_mu_t[o]    = sMuT[tid];
        }
        __syncthreads();

        // write covariance outputs
        {
            const size_t base = ((size_t)t * BB + b) * (DX * DX);
            for (int i = tid; i < DX * DX; i += NTHR) {
                out_Lam_next[base + i] = sLam[i];
                out_Lam_t[base + i]    = sLamT[i];
            }
        }
        __syncthreads();
    }
}

extern "C" void kernel_launch(void* const* d_in, const int* in_sizes, int n_in,
                              void* d_out, int out_size, void* d_ws, size_t ws_size,
                              hipStream_t stream) {
    (void)in_sizes; (void)n_in; (void)out_size; (void)d_ws; (void)ws_size;
    const float* a    = (const float*)d_in[0];
    const float* u    = (const float*)d_in[1];
    const float* mask = (const float*)d_in[2];
    const float* A    = (const float*)d_in[3];
    const float* Bm   = (const float*)d_in[4];
    const float* C    = (const float*)d_in[5];
    const float* mu0  = (const float*)d_in[6];
    const float* Lam0 = (const float*)d_in[7];
    const float* Wl   = (const float*)d_in[8];
    const float* Rl   = (const float*)d_in[9];
    float* out = (float*)d_out;

    const size_t smem_bytes = (size_t)SMEM_FLOATS * sizeof(float); // ~203 KB / WGP
    ldm_kalman_fwd<<<dim3(BB), dim3(NTHR), smem_bytes, stream>>>(
        a, u, mask, A, Bm, C, mu0, Lam0, Wl, Rl, out);
}